// TokenFuseLayer_82815559402205
// MI455X (gfx1250) — compile-verified
//
#include <hip/hip_runtime.h>

#define BDIM 8
#define NPIXC 16384
#define N0TOK 16384
#define N1TOK 4096
#define CH0 256
#define CH1 512
#define HM 128
#define WM 128
#define HD 64
#define WD 64

typedef __attribute__((ext_vector_type(2))) float v2f;
typedef __attribute__((ext_vector_type(8))) float v8f;

// ---------------------------------------------------------------------------
// Fused WMMA GEMM + BatchNorm (+ optional residual & ReLU).
// out[m,o] = epilogue( sum_k A[m,k] * Wt[o,k] )
// A: [M,KDIM] row-major; Wt: [NDIM,KDIM] row-major (weight rows = out channel)
// Block: 128 threads (4 waves), tile 16(M) x 64(N), K chunked by 64.
// Uses V_WMMA_F32_16X16X4_F32 (full fp32 — problem is memory-bound, no need
// to downcast; fp32 WMMA throughput >> the 17 GFLOP this workload needs).
// ---------------------------------------------------------------------------
template <int KDIM, int NDIM, bool RES_RELU>
__global__ __launch_bounds__(128) void wmma_gemm_bn(
    const float* __restrict__ A, const float* __restrict__ Wt,
    const float* __restrict__ bng, const float* __restrict__ bnb,
    const float* __restrict__ bnm, const float* __restrict__ bnv,
    const float* __restrict__ residual, float* __restrict__ out) {
  __shared__ float As[16][65];
  __shared__ float Bs[64][65];

  const int tid  = threadIdx.x;
  const int lane = tid & 31;
  const int wave = tid >> 5;
  const int half = lane >> 4;  // 0: lanes 0-15, 1: lanes 16-31
  const int ml   = lane & 15;
  const int m0   = blockIdx.x * 16;
  const int n0   = blockIdx.y * 64;
  const int nw   = wave * 16;  // this wave's 16-wide N slice inside the tile

  v8f acc = {0.f, 0.f, 0.f, 0.f, 0.f, 0.f, 0.f, 0.f};

  for (int k0 = 0; k0 < KDIM; k0 += 64) {
    // Stage A tile 16x64 (coalesced: contiguous K per row)
#pragma unroll
    for (int it = 0; it < 8; ++it) {
      int idx = tid + it * 128;
      int r = idx >> 6, c = idx & 63;
      As[r][c] = A[(size_t)(m0 + r) * KDIM + (k0 + c)];
    }
    // Stage B tile as Bs[k][n] = Wt[n0+n][k0+k] (coalesced reads over k)
#pragma unroll
    for (int it = 0; it < 32; ++it) {
      int idx = tid + it * 128;
      int n = idx >> 6, c = idx & 63;
      Bs[c][n] = Wt[(size_t)(n0 + n) * KDIM + (k0 + c)];
    }
    __syncthreads();

#pragma unroll
    for (int kk = 0; kk < 64; kk += 4) {
      // A 16x4 fp32 fragment: VGPR0 holds K = 2*half, VGPR1 holds K = 2*half+1
      v2f a, b;
      a.x = As[ml][kk + 2 * half];
      a.y = As[ml][kk + 2 * half + 1];
      // B 4x16 fp32 fragment: same half-wave K split, N striped across lanes
      b.x = Bs[kk + 2 * half][nw + ml];
      b.y = Bs[kk + 2 * half + 1][nw + ml];
      acc = __builtin_amdgcn_wmma_f32_16x16x4_f32(false, a, false, b,
                                                  (short)0, acc, false, false);
    }
    __syncthreads();
  }

  // Epilogue: BN over the output-channel (N) dim; one channel per lane.
  const int o = n0 + nw + ml;
  const float scale = bng[o] * rsqrtf(bnv[o] + 1e-5f);
  const float shift = bnb[o] - bnm[o] * scale;
#pragma unroll
  for (int r = 0; r < 8; ++r) {
    // C/D layout: VGPR r holds M = r (lanes 0-15) / r+8 (lanes 16-31)
    int mrow = m0 + r + 8 * half;
    float v = acc[r] * scale + shift;
    size_t oi = (size_t)mrow * NDIM + o;
    if (RES_RELU) v = fmaxf(residual[oi] + v, 0.f);
    out[oi] = v;
  }
}

// ---------------------------------------------------------------------------
// Pixel index per tcformer convention (jnp.round == round-half-even == rintf)
// ---------------------------------------------------------------------------
__device__ __forceinline__ int pix_idx(const float* __restrict__ loc2, int Hc,
                                       int Wc) {
  float x = fminf(fmaxf(loc2[0], -1.f), 1.f);
  float y = fminf(fmaxf(loc2[1], -1.f), 1.f);
  int px = (int)fminf(fmaxf(rintf((x + 1.f) * (float)Wc * 0.5f - 0.5f), 0.f),
                      (float)(Wc - 1));
  int py = (int)fminf(fmaxf(rintf((y + 1.f) * (float)Hc * 0.5f - 0.5f), 0.f),
                      (float)(Hc - 1));
  return py * Wc + px;
}

// token_downup(src[idx1] * w0) scattered into N0 tokens (out0 numerator) + den0
__global__ __launch_bounds__(CH0) void scatter_downup0(
    const float* __restrict__ src, const int* __restrict__ idx1,
    const int* __restrict__ idx0, const float* __restrict__ w0,
    float* __restrict__ acc, float* __restrict__ den) {
  int bp = blockIdx.x;
  int b = bp / NPIXC;
  int s = idx1[bp], t = idx0[bp];
  float w = w0[bp];
  int c = threadIdx.x;
  atomicAdd(&acc[((size_t)b * N0TOK + t) * CH0 + c],
            w * src[((size_t)b * N1TOK + s) * CH0 + c]);
  if (c == 0) atomicAdd(&den[b * N0TOK + t], w);
}

// out0 = relu(x0 + num/(den+eps)); num accumulated in-place in d_out
__global__ void finalize_out0(const float* __restrict__ x0,
                              const float* __restrict__ den,
                              float* __restrict__ out, size_t n) {
  for (size_t i = blockIdx.x * (size_t)blockDim.x + threadIdx.x; i < n;
       i += (size_t)gridDim.x * blockDim.x) {
    size_t tok = i / CH0;
    out[i] = fmaxf(x0[i] + out[i] / (den[tok] + 1e-6f), 0.f);
  }
}

// token2map: scatter x0[idx0] onto 128x128 map (NHWC) + hit count
__global__ __launch_bounds__(CH0) void scatter_token2map(
    const float* __restrict__ x0, const float* __restrict__ loc,
    const int* __restrict__ idx0, float* __restrict__ map,
    float* __restrict__ cnt) {
  int bp = blockIdx.x;
  int b = bp / NPIXC;
  int hw = pix_idx(&loc[(size_t)bp * 2], HM, WM);
  int t = idx0[bp];
  int c = threadIdx.x;
  atomicAdd(&map[((size_t)b * HM * WM + hw) * CH0 + c],
            x0[((size_t)b * N0TOK + t) * CH0 + c]);
  if (c == 0) atomicAdd(&cnt[b * HM * WM + hw], 1.f);
}

__global__ void map_avg(float* __restrict__ map, const float* __restrict__ cnt,
                        size_t n) {
  for (size_t i = blockIdx.x * (size_t)blockDim.x + threadIdx.x; i < n;
       i += (size_t)gridDim.x * blockDim.x) {
    map[i] /= (cnt[i / CH0] + 1e-6f);
  }
}

// depthwise 3x3 stride-2 pad-1 conv on NHWC map -> 64x64 NHWC
__global__ __launch_bounds__(CH0) void dwconv3x3s2(
    const float* __restrict__ map, const float* __restrict__ dw,
    float* __restrict__ outm) {
  int bq = blockIdx.x;
  int b = bq / (HD * WD);
  int q = bq % (HD * WD);
  int oy = q / WD, ox = q % WD;
  int c = threadIdx.x;
  float s = 0.f;
#pragma unroll
  for (int dy = 0; dy < 3; ++dy) {
    int iy = 2 * oy - 1 + dy;
    if (iy < 0 || iy >= HM) continue;
#pragma unroll
    for (int dx = 0; dx < 3; ++dx) {
      int ix = 2 * ox - 1 + dx;
      if (ix < 0 || ix >= WM) continue;
      s += map[((size_t)b * HM * WM + iy * WM + ix) * CH0 + c] *
           dw[c * 9 + dy * 3 + dx];
    }
  }
  outm[((size_t)b * HD * WD + q) * CH0 + c] = s;
}

// merged map2token + token_downup(x0) scatter into N1 tokens:
// acc += w1 * (x0[idx0]*skip + map_ds[pix64]);  den1 += w1
// (valid because both paths share den = segsum(w1 by idx1) and skip is linear)
__global__ __launch_bounds__(CH0) void scatter_down1(
    const float* __restrict__ x0, const float* __restrict__ mapds,
    const float* __restrict__ loc, const int* __restrict__ idx0,
    const int* __restrict__ idx1, const float* __restrict__ w1,
    const float* __restrict__ skip, float* __restrict__ acc,
    float* __restrict__ den) {
  int bp = blockIdx.x;
  int b = bp / NPIXC;
  int hw = pix_idx(&loc[(size_t)bp * 2], HD, WD);
  int s0 = idx0[bp], t1 = idx1[bp];
  float w = w1[bp];
  int c = threadIdx.x;
  float v = x0[((size_t)b * N0TOK + s0) * CH0 + c] * skip[c] +
            mapds[((size_t)b * HD * WD + hw) * CH0 + c];
  atomicAdd(&acc[((size_t)b * N1TOK + t1) * CH0 + c], w * v);
  if (c == 0) atomicAdd(&den[b * N1TOK + t1], w);
}

// in-place: acc = bn1(acc/(den+eps))
__global__ void finalize_bn1(float* __restrict__ acc,
                             const float* __restrict__ den,
                             const float* __restrict__ g,
                             const float* __restrict__ bb,
                             const float* __restrict__ mm,
                             const float* __restrict__ vv, size_t n) {
  for (size_t i = blockIdx.x * (size_t)blockDim.x + threadIdx.x; i < n;
       i += (size_t)gridDim.x * blockDim.x) {
    size_t tok = i / CH0;
    int c = (int)(i % CH0);
    float x = acc[i] / (den[tok] + 1e-6f);
    float sc = g[c] * rsqrtf(vv[c] + 1e-5f);
    acc[i] = (x - mm[c]) * sc + bb[c];
  }
}

extern "C" void kernel_launch(void* const* d_in, const int* in_sizes, int n_in,
                              void* d_out, int out_size, void* d_ws,
                              size_t ws_size, hipStream_t stream) {
  (void)in_sizes; (void)n_in; (void)out_size; (void)ws_size;
  const float* x0   = (const float*)d_in[0];
  const float* x1   = (const float*)d_in[1];
  const int*   idx0 = (const int*)d_in[2];
  const int*   idx1 = (const int*)d_in[3];
  const float* w0   = (const float*)d_in[4];
  const float* w1   = (const float*)d_in[5];
  const float* loc  = (const float*)d_in[6];
  const float* W01  = (const float*)d_in[7];
  const float* bn01g = (const float*)d_in[8];
  const float* bn01b = (const float*)d_in[9];
  const float* bn01m = (const float*)d_in[10];
  const float* bn01v = (const float*)d_in[11];
  const float* dww   = (const float*)d_in[12];
  const float* skip  = (const float*)d_in[13];
  const float* bn1g  = (const float*)d_in[14];
  const float* bn1b  = (const float*)d_in[15];
  const float* bn1m  = (const float*)d_in[16];
  const float* bn1v  = (const float*)d_in[17];
  const float* convw = (const float*)d_in[18];
  const float* bn2g  = (const float*)d_in[19];
  const float* bn2b  = (const float*)d_in[20];
  const float* bn2m  = (const float*)d_in[21];
  const float* bn2v  = (const float*)d_in[22];
  // d_in[23]/[24] are H0/W0 (=128), baked in as compile-time constants.

  float* out0 = (float*)d_out;                       // [B,N0,C0] (acc in place)
  float* out1 = out0 + (size_t)BDIM * N0TOK * CH0;   // [B,N1,C1]

  float* ws    = (float*)d_ws;
  float* src   = ws;                                        // B*N1*C0
  float* map   = src + (size_t)BDIM * N1TOK * CH0;          // B*HM*WM*C0 (NHWC)
  float* cnt   = map + (size_t)BDIM * HM * WM * CH0;        // B*HM*WM
  float* mapds = cnt + (size_t)BDIM * HM * WM;              // B*HD*WD*C0
  float* acc1  = mapds + (size_t)BDIM * HD * WD * CH0;      // B*N1*C0
  float* den1  = acc1 + (size_t)BDIM * N1TOK * CH0;         // B*N1
  float* den0  = den1 + (size_t)BDIM * N1TOK;               // B*N0

  // Zero accumulators (graph-capturable async memsets).
  (void)hipMemsetAsync(out0, 0, (size_t)BDIM * N0TOK * CH0 * sizeof(float),
                       stream);
  (void)hipMemsetAsync(
      map, 0, ((size_t)BDIM * HM * WM * CH0 + (size_t)BDIM * HM * WM) *
                  sizeof(float), stream);
  (void)hipMemsetAsync(
      acc1, 0,
      ((size_t)BDIM * N1TOK * CH0 + (size_t)BDIM * N1TOK +
       (size_t)BDIM * N0TOK) * sizeof(float), stream);

  // --- branch 0: src = bn01(x1 @ W01^T)  (M=32768,K=512,N=256) ---
  dim3 g1(BDIM * N1TOK / 16, CH0 / 64);
  wmma_gemm_bn<CH1, CH0, false><<<g1, 128, 0, stream>>>(
      x1, W01, bn01g, bn01b, bn01m, bn01v, nullptr, src);

  scatter_downup0<<<BDIM * NPIXC, CH0, 0, stream>>>(src, idx1, idx0, w0, out0,
                                                    den0);
  finalize_out0<<<2048, 256, 0, stream>>>(x0, den0, out0,
                                          (size_t)BDIM * N0TOK * CH0);

  // --- branch 1: token2map -> dwconv s2 -> map2token (+ skip downup merged) ---
  scatter_token2map<<<BDIM * NPIXC, CH0, 0, stream>>>(x0, loc, idx0, map, cnt);
  map_avg<<<2048, 256, 0, stream>>>(map, cnt, (size_t)BDIM * HM * WM * CH0);
  dwconv3x3s2<<<BDIM * HD * WD, CH0, 0, stream>>>(map, dww, mapds);
  scatter_down1<<<BDIM * NPIXC, CH0, 0, stream>>>(x0, mapds, loc, idx0, idx1,
                                                  w1, skip, acc1, den1);
  finalize_bn1<<<2048, 256, 0, stream>>>(acc1, den1, bn1g, bn1b, bn1m, bn1v,
                                         (size_t)BDIM * N1TOK * CH0);

  // --- out1 = relu(x1 + bn2(acc1 @ conv_w^T))  (M=32768,K=256,N=512) ---
  dim3 g2(BDIM * N1TOK / 16, CH1 / 64);
  wmma_gemm_bn<CH0, CH1, true><<<g2, 128, 0, stream>>>(
      acc1, convw, bn2g, bn2b, bn2m, bn2v, x1, out1);
}